// GCN_4337916969345
// MI455X (gfx1250) — compile-verified
//
#include <hip/hip_runtime.h>

#define N_NODES 170000
#define N_EDGES 1200000
#define BN_EPS  1e-5f

typedef __attribute__((ext_vector_type(2))) float v2f;
typedef __attribute__((ext_vector_type(8))) float v8f;

__device__ __forceinline__ void atomAddF(float* p, float v) {
  unsafeAtomicAdd(p, v);   // global_atomic_add_f32 on gfx1250
}

// ---------------- degree / normalization ----------------
__global__ void init_deg(float* __restrict__ deg) {
  int i = blockIdx.x * blockDim.x + threadIdx.x;
  if (i < N_NODES) deg[i] = 1.0f;   // self-loop
}

__global__ void edge_deg(const long long* __restrict__ ei, float* __restrict__ deg) {
  int e = blockIdx.x * blockDim.x + threadIdx.x;
  if (e < N_EDGES) {
    int d = (int)ei[(size_t)N_EDGES + e];   // row 1 = dst
    atomAddF(&deg[d], 1.0f);
  }
}

__global__ void finalize_dinv(float* __restrict__ deg) {
  int i = blockIdx.x * blockDim.x + threadIdx.x;
  if (i < N_NODES) {
    float d = deg[i];
    deg[i] = (d > 0.0f) ? rsqrtf(d) : 0.0f;
  }
}

// ---------------- dense GEMM: H = X(N x 128) @ W(128 x COUT), f32 WMMA ----------
// block = 256 threads (8 waves); each block computes a 128-row stripe, full width.
// W staged in LDS with k-pairs interleaved so a B fragment is ONE ds_load_b64:
//   sW[((k>>1)*COUT_PAD + n)*2 + (k&1)] = W[k][n]
template<int COUT, int COUT_PAD>
__global__ __launch_bounds__(256) void gemm_wmma(const float* __restrict__ X,
                                                 const float* __restrict__ W,
                                                 float* __restrict__ H) {
  constexpr int NT = COUT_PAD / 16;
  __shared__ float sW[128 * COUT_PAD];

  const int tid = threadIdx.x;
  for (int i = tid; i < 128 * COUT_PAD; i += 256) {
    int k = i / COUT_PAD, n = i % COUT_PAD;
    float v = (n < COUT) ? W[k * COUT + n] : 0.0f;
    sW[((k >> 1) * COUT_PAD + n) * 2 + (k & 1)] = v;
  }
  __syncthreads();

  const int wave  = tid >> 5;
  const int lane  = tid & 31;
  const int m     = lane & 15;   // row within A tile / col within B,C tile
  const int khalf = lane >> 4;   // selects K pair {0,1} vs {2,3}
  const int row   = blockIdx.x * 128 + wave * 16 + m;
  // clamp OOB rows to a valid address: their D rows are never stored
  const int rowc  = (row < N_NODES) ? row : (N_NODES - 1);

  v8f acc[NT];
  #pragma unroll
  for (int t = 0; t < NT; ++t)
    #pragma unroll
    for (int i = 0; i < 8; ++i) acc[t][i] = 0.0f;

  const float* xrow = X + (size_t)rowc * 128;

  #pragma unroll 4
  for (int kk = 0; kk < 32; ++kk) {
    const int k0 = kk * 4 + khalf * 2;          // even
    const int kp = kk * 2 + khalf;              // k-pair index
    // A fragment (16x4 f32): lane m holds (K=k0, K=k0+1) in x,y
    v2f a = *reinterpret_cast<const v2f*>(xrow + k0);
    #pragma unroll
    for (int t = 0; t < NT; ++t) {
      // B fragment (4x16 f32): single b64 LDS load, rows k0,k0+1 at col n
      v2f b = *reinterpret_cast<const v2f*>(sW + (size_t)(kp * COUT_PAD + t * 16 + m) * 2);
      acc[t] = __builtin_amdgcn_wmma_f32_16x16x4_f32(
          false, a, false, b, (short)0, acc[t], false, false);
    }
  }

  // D layout: VGPR r -> row (wave*16 + khalf*8 + r), col t*16 + m
  const int rbase = blockIdx.x * 128 + wave * 16 + khalf * 8;
  #pragma unroll
  for (int t = 0; t < NT; ++t) {
    const int n = t * 16 + m;
    if (n < COUT) {
      #pragma unroll
      for (int r = 0; r < 8; ++r) {
        const int orow = rbase + r;
        if (orow < N_NODES) H[(size_t)orow * COUT + n] = acc[t][r];
      }
    }
  }
}

// ---------------- aggregation ----------------
// out[i][c] = dinv[i]^2 * H[i][c] + bias[c]   (self-loop + bias), float4 per thread
template<int C>
__global__ void self_bias_init(const float* __restrict__ H, const float* __restrict__ dinv,
                               const float* __restrict__ bias, float* __restrict__ out) {
  constexpr int C4 = C / 4;
  size_t idx = (size_t)blockIdx.x * blockDim.x + threadIdx.x;
  if (idx >= (size_t)N_NODES * C4) return;
  int row = (int)(idx / C4);
  int c4  = (int)(idx % C4) * 4;
  float di = dinv[row];
  float w  = di * di;
  float4 h = *reinterpret_cast<const float4*>(H + (size_t)row * C + c4);
  float4 b = *reinterpret_cast<const float4*>(bias + c4);
  float4 o;
  o.x = w * h.x + b.x; o.y = w * h.y + b.y;
  o.z = w * h.z + b.z; o.w = w * h.w + b.w;
  *reinterpret_cast<float4*>(out + (size_t)row * C + c4) = o;
}

// TPE threads per edge (power of two); each active thread handles 4 channels.
template<int C, int TPE>
__global__ void edge_scatter(const long long* __restrict__ ei, const float* __restrict__ H,
                             const float* __restrict__ dinv, float* __restrict__ out) {
  size_t t = (size_t)blockIdx.x * blockDim.x + threadIdx.x;
  size_t e = t / TPE;
  int   c4 = (int)(t % TPE) * 4;
  if (e >= N_EDGES) return;
  int s = (int)ei[e];
  int d = (int)ei[(size_t)N_EDGES + e];
  if (c4 < C) {
    float w = dinv[s] * dinv[d];
    float4 h = *reinterpret_cast<const float4*>(H + (size_t)s * C + c4);
    float* o = out + (size_t)d * C + c4;
    atomAddF(o + 0, w * h.x);
    atomAddF(o + 1, w * h.y);
    atomAddF(o + 2, w * h.z);
    atomAddF(o + 3, w * h.w);
  }
}

// ---------------- batchnorm ----------------
__global__ void zero_stats(float* __restrict__ stats) {
  if (threadIdx.x < 256) stats[threadIdx.x] = 0.0f;
}

__global__ __launch_bounds__(128) void bn_stats(const float* __restrict__ h,
                                                float* __restrict__ stats) {
  const int c = threadIdx.x;
  size_t r0 = (size_t)blockIdx.x * 256;
  size_t r1 = r0 + 256; if (r1 > N_NODES) r1 = N_NODES;
  float s = 0.0f, s2 = 0.0f;
  for (size_t r = r0; r < r1; ++r) {
    float v = h[r * 128 + c];
    s += v; s2 += v * v;
  }
  atomAddF(&stats[c], s);
  atomAddF(&stats[128 + c], s2);
}

__global__ __launch_bounds__(128) void bn_apply(float* __restrict__ h,
                                                const float* __restrict__ stats,
                                                const float* __restrict__ g,
                                                const float* __restrict__ be) {
  const int c = threadIdx.x;
  const float inv_n = 1.0f / (float)N_NODES;
  float mean = stats[c] * inv_n;
  float var  = stats[128 + c] * inv_n - mean * mean;
  float sc   = rsqrtf(var + BN_EPS) * g[c];
  float sh   = be[c] - mean * sc;
  size_t r0 = (size_t)blockIdx.x * 256;
  size_t r1 = r0 + 256; if (r1 > N_NODES) r1 = N_NODES;
  for (size_t r = r0; r < r1; ++r) {
    float v = h[r * 128 + c] * sc + sh;
    h[r * 128 + c] = fmaxf(v, 0.0f);
  }
}

// ---------------- driver ----------------
extern "C" void kernel_launch(void* const* d_in, const int* in_sizes, int n_in,
                              void* d_out, int out_size, void* d_ws, size_t ws_size,
                              hipStream_t stream) {
  (void)in_sizes; (void)n_in; (void)out_size; (void)ws_size;
  const float*     x   = (const float*)d_in[0];
  const long long* ei  = (const long long*)d_in[1];
  const float*     W0  = (const float*)d_in[2];
  const float*     b0  = (const float*)d_in[3];
  const float*     g0  = (const float*)d_in[4];
  const float*     be0 = (const float*)d_in[5];
  const float*     W1  = (const float*)d_in[6];
  const float*     b1  = (const float*)d_in[7];
  const float*     g1  = (const float*)d_in[8];
  const float*     be1 = (const float*)d_in[9];
  const float*     W2  = (const float*)d_in[10];
  const float*     b2  = (const float*)d_in[11];
  float* out = (float*)d_out;

  float* ws    = (float*)d_ws;
  float* dinv  = ws;                                   // N
  float* H     = ws + N_NODES;                         // N*128
  float* AGG   = H  + (size_t)N_NODES * 128;           // N*128
  float* stats = AGG + (size_t)N_NODES * 128;          // 256

  const int T = 256;
  const int nb_nodes = (N_NODES + T - 1) / T;
  const int nb_edges = (N_EDGES + T - 1) / T;
  const int gemm_blocks = (N_NODES + 127) / 128;
  const int sb_blocks_128 = (int)(((size_t)N_NODES * 32 + T - 1) / T);   // C4=32
  const int sb_blocks_40  = (int)(((size_t)N_NODES * 10 + T - 1) / T);   // C4=10
  const int es_blocks_128 = (int)(((size_t)N_EDGES * 32 + T - 1) / T);   // TPE=32
  const int es_blocks_40  = (int)(((size_t)N_EDGES * 16 + T - 1) / T);   // TPE=16
  const int bn_blocks = (N_NODES + 255) / 256;

  // normalization coefficients
  init_deg<<<nb_nodes, T, 0, stream>>>(dinv);
  edge_deg<<<nb_edges, T, 0, stream>>>(ei, dinv);
  finalize_dinv<<<nb_nodes, T, 0, stream>>>(dinv);

  // layer 0: conv + BN + ReLU
  gemm_wmma<128, 128><<<gemm_blocks, 256, 0, stream>>>(x, W0, H);
  self_bias_init<128><<<sb_blocks_128, T, 0, stream>>>(H, dinv, b0, AGG);
  edge_scatter<128, 32><<<es_blocks_128, T, 0, stream>>>(ei, H, dinv, AGG);
  zero_stats<<<1, 256, 0, stream>>>(stats);
  bn_stats<<<bn_blocks, 128, 0, stream>>>(AGG, stats);
  bn_apply<<<bn_blocks, 128, 0, stream>>>(AGG, stats, g0, be0);

  // layer 1: conv + BN + ReLU
  gemm_wmma<128, 128><<<gemm_blocks, 256, 0, stream>>>(AGG, W1, H);
  self_bias_init<128><<<sb_blocks_128, T, 0, stream>>>(H, dinv, b1, AGG);
  edge_scatter<128, 32><<<es_blocks_128, T, 0, stream>>>(ei, H, dinv, AGG);
  zero_stats<<<1, 256, 0, stream>>>(stats);
  bn_stats<<<bn_blocks, 128, 0, stream>>>(AGG, stats);
  bn_apply<<<bn_blocks, 128, 0, stream>>>(AGG, stats, g1, be1);

  // layer 2: conv -> d_out
  gemm_wmma<40, 48><<<gemm_blocks, 256, 0, stream>>>(AGG, W2, H);
  self_bias_init<40><<<sb_blocks_40, T, 0, stream>>>(H, dinv, b2, out);
  edge_scatter<40, 16><<<es_blocks_40, T, 0, stream>>>(ei, H, dinv, out);
}